// ROINetwork_60670708023372
// MI455X (gfx1250) — compile-verified
//
#include <hip/hip_runtime.h>

typedef __attribute__((ext_vector_type(2))) float v2f;
typedef __attribute__((ext_vector_type(8))) float v8f;

#if defined(__HIP_DEVICE_COMPILE__) && __has_builtin(__builtin_amdgcn_wmma_f32_16x16x4_f32)
#define HAS_WMMA4 1
#else
#define HAS_WMMA4 0
#endif

#if defined(__HIP_DEVICE_COMPILE__) && __has_builtin(__builtin_amdgcn_global_load_async_to_lds_b32)
#define HAS_ASYNC_LD 1
#else
#define HAS_ASYNC_LD 0
#endif

#if defined(__HIP_DEVICE_COMPILE__) && __has_builtin(__builtin_amdgcn_global_store_async_from_lds_b32)
#define HAS_ASYNC_ST 1
#else
#define HAS_ASYNC_ST 0
#endif

#if defined(__HIP_DEVICE_COMPILE__)
__device__ __forceinline__ void wait_asynccnt0() {
#if __has_builtin(__builtin_amdgcn_s_wait_asynccnt)
    __builtin_amdgcn_s_wait_asynccnt(0);
#else
    asm volatile("s_wait_asynccnt 0x0" ::: "memory");
#endif
}
#else
__device__ __forceinline__ void wait_asynccnt0() {}
#endif

// async-LDS builtins take (global int* AS1, lds int* AS3, imm offset, imm cpol)
#define AS1P(p) ((__attribute__((address_space(1))) int*)(p))
#define AS3P(p) ((__attribute__((address_space(3))) int*)(p))

// Separable ROI-Align:
//   out[c,ph,pw] = sum_j sum_i Wy[ph,j] * Wx[pw,i] * patch[c,j,i]
// One block per ROI. 8 waves (wave32). Channels in 8 chunks of 32.
// Stage 1 (WMMA f32 16x16x4): T[(ch,j),pw] = sum_i patch[(ch,j),i] * Wx[pw,i]
// Stage 2 (WMMA f32 16x16x4): out[(ch,pw),ph] = sum_j T[(ch,j),pw] * Wy[ph,j]

template <int MAXD>
struct SmemL {
    static constexpr int RS       = MAXD + 1;          // odd patch row stride
    static constexpr int WSTR     = MAXD + 1;          // odd weight row stride
    static constexpr int PATCH    = 0;
    static constexpr int PATCH_SZ = 32 * MAXD * RS;
    static constexpr int T_OFF    = PATCH_SZ;          // T stride 9 (odd)
    static constexpr int T_SZ     = 32 * MAXD * 9;
    static constexpr int WX       = T_OFF + T_SZ;      // 16 x WSTR (rows 7..15 zero)
    static constexpr int WY       = WX + 16 * WSTR;
    static constexpr int OB       = WY + 16 * WSTR;    // 32 ch x 49 bins
    static constexpr int TOTAL    = OB + 32 * 49;
};

template <int MAXD>
__launch_bounds__(256, 1)
__global__ void roi_align_wmma(const float* __restrict__ feat,
                               const float* __restrict__ boxes,
                               float* __restrict__ out,
                               int H, int W, float scale)
{
    using L = SmemL<MAXD>;
    extern __shared__ float smem[];

    const int tid   = threadIdx.x;
    const int wave  = tid >> 5;
    const int lane  = tid & 31;
    const int lrow  = lane & 15;   // WMMA: matrix row / B column index
    const int khalf = lane >> 4;   // WMMA: which K pair this lane holds

    const int    k  = blockIdx.x;
    const float* bx = boxes + (size_t)k * 5;
    const int  bidx = (int)bx[0];
    const float x1  = bx[1] * scale;
    const float y1  = bx[2] * scale;
    const float x2  = bx[3] * scale;
    const float y2  = bx[4] * scale;
    const float bw  = fmaxf(x2 - x1, 1.0f) / 7.0f;
    const float bh  = fmaxf(y2 - y1, 1.0f) / 7.0f;

    // patch origin = floor of first (clipped) sample coordinate (samples are monotonic)
    const int x0 = (int)floorf(fminf(fmaxf(x1 + bw * 0.125f, 0.0f), (float)(W - 1)));
    const int y0 = (int)floorf(fminf(fmaxf(y1 + bh * 0.125f, 0.0f), (float)(H - 1)));

    // zero Wx and Wy (contiguous 2 * 16 * WSTR floats)
    for (int e = tid; e < 2 * 16 * L::WSTR; e += 256) smem[L::WX + e] = 0.0f;
    __syncthreads();

    // Build separable weight matrices. 14 threads, each owning one (axis, bin):
    // no races since one thread accumulates all 4 sub-samples of its bin row.
    if (tid < 14) {
        const bool  isy  = tid >= 7;
        const int   p    = isy ? tid - 7 : tid;
        const float st   = isy ? y1 : x1;
        const float bsz  = isy ? bh : bw;
        const float size = isy ? (float)H : (float)W;
        const int   org  = isy ? y0 : x0;
        float* Wm = smem + (isy ? L::WY : L::WX) + p * L::WSTR;
        for (int ss = 0; ss < 4; ++ss) {
            const float coord = st + (float)p * bsz + ((float)ss + 0.5f) * bsz * 0.25f;
            const float vf = (coord > -1.0f && coord < size) ? 0.25f : 0.0f; // 1/4 avg folded
            const float c  = fminf(fmaxf(coord, 0.0f), size - 1.0f);
            const int   lo = (int)floorf(c);
            const int   hi = min(lo + 1, (int)size - 1);
            const float fr = c - (float)lo;
            const int   il = min(max(lo - org, 0), MAXD - 1);
            const int   ih = min(max(hi - org, 0), MAXD - 1);
            Wm[il] += (1.0f - fr) * vf;
            Wm[ih] += fr * vf;
        }
    }
    __syncthreads();

    const size_t HW    = (size_t)H * (size_t)W;
    const float* fbase = feat + (size_t)bidx * 256 * HW;

    for (int cc = 0; cc < 8; ++cc) {
        const float* fb = fbase + (size_t)(cc * 32) * HW;

        // ---- stage contiguous patch rows into LDS (async DMA path on CDNA5)
        for (int e = tid; e < 32 * MAXD * MAXD; e += 256) {
            const int i  = e % MAXD;
            const int rj = e / MAXD;
            const int j  = rj % MAXD;
            const int ch = rj / MAXD;
            const int gy = min(y0 + j, H - 1);   // padded rows read clamped (weight 0)
            const int gx = min(x0 + i, W - 1);
            const float* src = &fb[(size_t)ch * HW + (size_t)gy * W + gx];
            float*       dst = &smem[L::PATCH + (ch * MAXD + j) * L::RS + i];
#if HAS_ASYNC_LD
            __builtin_amdgcn_global_load_async_to_lds_b32(AS1P(src), AS3P(dst), 0, 0);
#else
            *dst = *src;
#endif
        }
        // warm next chunk in L2 while we compute
        if (cc < 7 && tid < 32) {
            __builtin_prefetch(fbase + (size_t)((cc + 1) * 32 + tid) * HW +
                                   (size_t)y0 * W + x0, 0, 1);
        }
#if HAS_ASYNC_LD || HAS_ASYNC_ST
        // covers this chunk's patch loads and the previous chunk's OB stores
        wait_asynccnt0();
#endif
        __syncthreads();

        // ---- stage 1: T[(ch,j), pw] = sum_i patch * Wx[pw,i]   (M = 32*MAXD rows)
        for (int t = wave; t < 2 * MAXD; t += 8) {
            v8f acc = {};
            const int    arow = t * 16 + lrow;
            const float* ap   = &smem[L::PATCH + arow * L::RS];
            const float* bp   = &smem[L::WX + lrow * L::WSTR];
#if HAS_WMMA4
#pragma unroll
            for (int kt = 0; kt < MAXD / 4; ++kt) {
                const int kk = kt * 4 + khalf * 2; // lanes 0-15: K=0,1 ; lanes 16-31: K=2,3
                v2f a; a.x = ap[kk]; a.y = ap[kk + 1];
                v2f b; b.x = bp[kk]; b.y = bp[kk + 1];
                acc = __builtin_amdgcn_wmma_f32_16x16x4_f32(
                    false, a, false, b, (short)0, acc, false, false);
            }
#else
#pragma unroll
            for (int g = 0; g < 8; ++g) {
                const int row = t * 16 + g + 8 * khalf;
                float s = 0.0f;
                for (int kk = 0; kk < MAXD; ++kk)
                    s += smem[L::PATCH + row * L::RS + kk] *
                         smem[L::WX + lrow * L::WSTR + kk];
                acc[g] = s;
            }
#endif
            if (lrow < 8) {           // only cols pw = 0..7 are meaningful (7 used)
#pragma unroll
                for (int g = 0; g < 8; ++g) {
                    const int m = g + 8 * khalf;   // D row for this lane/vgpr
                    smem[L::T_OFF + (t * 16 + m) * 9 + lrow] = acc[g];
                }
            }
        }
        __syncthreads();

        // ---- stage 2: out[(ch,pw), ph] = sum_j T[(ch,j), pw] * Wy[ph,j]  (M = 256 rows)
        for (int t = wave; t < 16; t += 8) {
            v8f acc = {};
            const int arow = t * 16 + lrow;    // = ch*8 + pw
            const int ch   = arow >> 3;
            const int pw   = arow & 7;
            const float* bp = &smem[L::WY + lrow * L::WSTR];
#if HAS_WMMA4
#pragma unroll
            for (int kt = 0; kt < MAXD / 4; ++kt) {
                const int kk = kt * 4 + khalf * 2;
                v2f a;
                a.x = smem[L::T_OFF + (ch * MAXD + kk) * 9 + pw];
                a.y = smem[L::T_OFF + (ch * MAXD + kk + 1) * 9 + pw];
                v2f b; b.x = bp[kk]; b.y = bp[kk + 1];
                acc = __builtin_amdgcn_wmma_f32_16x16x4_f32(
                    false, a, false, b, (short)0, acc, false, false);
            }
#else
#pragma unroll
            for (int g = 0; g < 8; ++g) {
                const int row = t * 16 + g + 8 * khalf;
                const int c2 = row >> 3, p2 = row & 7;
                float s = 0.0f;
                for (int kk = 0; kk < MAXD; ++kk)
                    s += smem[L::T_OFF + (c2 * MAXD + kk) * 9 + p2] *
                         smem[L::WY + lrow * L::WSTR + kk];
                acc[g] = s;
            }
#endif
            if (lrow < 7) {
                const int ph = lrow;
#pragma unroll
                for (int g = 0; g < 8; ++g) {
                    const int m   = g + 8 * khalf;
                    const int row = t * 16 + m;
                    const int och = row >> 3;
                    const int opw = row & 7;
                    if (opw < 7)
                        smem[L::OB + och * 49 + ph * 7 + opw] = acc[g];
                }
            }
        }
        __syncthreads();

        // ---- fully coalesced store of this 32-channel chunk (async LDS->global)
        float* ob = out + ((size_t)k * 256 + (size_t)cc * 32) * 49;
        for (int e = tid; e < 32 * 49; e += 256) {
#if HAS_ASYNC_ST
            __builtin_amdgcn_global_store_async_from_lds_b32(
                AS1P(&ob[e]), AS3P(&smem[L::OB + e]), 0, 0);
#else
            ob[e] = smem[L::OB + e];
#endif
        }
        __syncthreads();
        // pending async OB stores are drained by the next chunk's wait_asynccnt0()
        // (before OB is rewritten) or by the implicit S_WAIT_IDLE at S_ENDPGM.
    }
}

extern "C" void kernel_launch(void* const* d_in, const int* in_sizes, int n_in,
                              void* d_out, int out_size, void* d_ws, size_t ws_size,
                              hipStream_t stream)
{
    (void)in_sizes; (void)n_in; (void)d_ws; (void)ws_size; (void)out_size;
    const float* f2 = (const float*)d_in[0];
    const float* f3 = (const float*)d_in[1];
    const float* f4 = (const float*)d_in[2];
    const float* f5 = (const float*)d_in[3];
    const float* b2 = (const float*)d_in[4];
    const float* b3 = (const float*)d_in[5];
    const float* b4 = (const float*)d_in[6];
    const float* b5 = (const float*)d_in[7];
    float* out = (float*)d_out;

    size_t o = 0;
    roi_align_wmma<28><<<1024, 256, SmemL<28>::TOTAL * sizeof(float), stream>>>(
        f2, b2, out + o, 200, 200, 200.0f / 800.0f);
    o += (size_t)1024 * 256 * 49;
    roi_align_wmma<16><<<1024, 256, SmemL<16>::TOTAL * sizeof(float), stream>>>(
        f3, b3, out + o, 100, 100, 100.0f / 800.0f);
    o += (size_t)1024 * 256 * 49;
    roi_align_wmma<12><<<512, 256, SmemL<12>::TOTAL * sizeof(float), stream>>>(
        f4, b4, out + o, 50, 50, 50.0f / 800.0f);
    o += (size_t)512 * 256 * 49;
    roi_align_wmma<8><<<256, 256, SmemL<8>::TOTAL * sizeof(float), stream>>>(
        f5, b5, out + o, 25, 25, 25.0f / 800.0f);
}